// SimpleMambaModel_42769284333960
// MI455X (gfx1250) — compile-verified
//
#include <hip/hip_runtime.h>
#include <hip/hip_bf16.h>

typedef float v2f __attribute__((ext_vector_type(2)));
typedef float v8f __attribute__((ext_vector_type(8)));

// Model dims (fixed by the reference harness)
#define DMODEL  1024
#define DINNER  2048
#define DSTATE  16
#define DTRANK  64
#define SEQL    1024
#define BSZ     2
#define NLAYER  2

// ---------------------------------------------------------------------------
// Generic fp32 WMMA GEMM:  C[M,N'] = act( A[M,K](lda) * W[N',K](ldw)^T + bias )
// Tiling: each wave computes a 32 x (JT*16) tile (2 M-tiles x JT N-tiles),
// and all 8 waves of a block share the same N-strip so the B (weight)
// fragments hit the WGP cache. JT is compile-time -> branch-free inner loop;
// k-loop unrolled x2 with all loads issued before the WMMAs.
// act: 0 = none, 1 = softplus
// ---------------------------------------------------------------------------
template<int JT>
__global__ __launch_bounds__(256) void wmma_gemm_f32(
    const float* __restrict__ A, long lda,
    const float* __restrict__ W, long ldw,
    float* __restrict__ C, long ldc,
    const float* __restrict__ bias,
    int mtiles, int nstrips, int K, int act)
{
    // all 8 waves of this block share ns (same B strip), span M
    const int ns = blockIdx.x % nstrips;
    const int mg = blockIdx.x / nstrips;
    // scalarize wave id: tile selection uniform -> EXEC all-1s at every WMMA
    const int w  = __builtin_amdgcn_readfirstlane(threadIdx.x >> 5);
    const int lane = threadIdx.x & 31;

    const int mpair = mg * 8 + w;          // 2 M-tiles (32 rows) per wave
    if (mpair * 2 >= mtiles) return;
    const int m0 = mpair << 5;

    const int half = lane >> 4;   // 0: k+{0,1}, 1: k+{2,3}
    const int lr   = lane & 15;

    // A fragment bases: rows m0+lr and m0+16+lr, column offset 2*half
    const float* Arow0 = A + (size_t)(m0 + lr) * lda + (half << 1);
    const float* Arow1 = Arow0 + (size_t)16 * lda;

    // B fragment bases: row n = ns*JT*16 + j*16 + lr of W, col offset 2*half
    const float* Wb[JT];
    #pragma unroll
    for (int j = 0; j < JT; ++j)
        Wb[j] = W + (size_t)((ns * JT + j) * 16 + lr) * ldw + (half << 1);

    v8f acc[2][JT];
    #pragma unroll
    for (int mi = 0; mi < 2; ++mi)
        #pragma unroll
        for (int j = 0; j < JT; ++j)
            #pragma unroll
            for (int v = 0; v < 8; ++v) acc[mi][j][v] = 0.f;

    // K is a multiple of 8 for every GEMM in this model (1024/2048/64)
    for (int k0 = 0; k0 < K; k0 += 8) {
        v2f a0[2], a1[2];
        v2f bv[2][JT];
        #pragma unroll
        for (int u = 0; u < 2; ++u) {
            a0[u] = *(const v2f*)(Arow0 + k0 + 4 * u);
            a1[u] = *(const v2f*)(Arow1 + k0 + 4 * u);
            #pragma unroll
            for (int j = 0; j < JT; ++j)
                bv[u][j] = *(const v2f*)(Wb[j] + k0 + 4 * u);
        }
        #pragma unroll
        for (int u = 0; u < 2; ++u) {
            #pragma unroll
            for (int j = 0; j < JT; ++j) {
                acc[0][j] = __builtin_amdgcn_wmma_f32_16x16x4_f32(
                                false, a0[u], false, bv[u][j],
                                (short)0, acc[0][j], false, false);
                acc[1][j] = __builtin_amdgcn_wmma_f32_16x16x4_f32(
                                false, a1[u], false, bv[u][j],
                                (short)0, acc[1][j], false, false);
            }
        }
    }

    // Epilogue. C/D layout: n = lane&15, m = tile_m0 + 8*half + vgpr
    #pragma unroll
    for (int mi = 0; mi < 2; ++mi) {
        #pragma unroll
        for (int j = 0; j < JT; ++j) {
            const int n = (ns * JT + j) * 16 + lr;
            const float bvad = bias ? bias[n] : 0.f;
            #pragma unroll
            for (int v = 0; v < 8; ++v) {
                const int m = m0 + (mi << 4) + (half << 3) + v;
                float val = acc[mi][j][v] + bvad;
                if (act == 1)   // softplus
                    val = (val > 20.f) ? val : log1pf(expf(val));
                C[(size_t)m * ldc + n] = val;
            }
        }
    }
}

// ---------------------------------------------------------------------------
// Causal depthwise conv1d (d_conv=4) + bias + SiLU.
// Reads xr = xz[:, 0:DINNER] (row stride 2*DINNER), writes xc (B*L, DINNER).
// ---------------------------------------------------------------------------
__global__ __launch_bounds__(256) void conv_silu_kernel(
    const float* __restrict__ xz,
    const float* __restrict__ cw,    // (DINNER, 4)
    const float* __restrict__ cb,    // (DINNER)
    float* __restrict__ xc)
{
    const int idx = blockIdx.x * blockDim.x + threadIdx.x;
    const int total = BSZ * SEQL * DINNER;
    if (idx >= total) return;
    const int d = idx & (DINNER - 1);
    const int t = idx >> 11;              // b*L + l   (DINNER = 2048 = 2^11)
    const int l = t & (SEQL - 1);

    float sum = cb[d];
    const float* w = cw + d * 4;
    #pragma unroll
    for (int k = 0; k < 4; ++k) {
        const int ll = l - 3 + k;
        if (ll >= 0)
            sum += xz[(size_t)(t - 3 + k) * (2 * DINNER) + d] * w[k];
    }
    xc[idx] = sum / (1.f + expf(-sum));   // silu
}

// ---------------------------------------------------------------------------
// Selective scan. One 16-lane group per (b, d) channel: lane s holds state s.
// h = exp(dt*A)*h + dt*B*xc ;  y = <h, C> + xc*D ;  y *= silu(z)
// y is written into the dead xr half of xz (row stride 4096) for out_proj.
// ---------------------------------------------------------------------------
__global__ __launch_bounds__(256) void mamba_scan_kernel(
    const float* __restrict__ dt,     // (B*L, DINNER)  post-softplus
    const float* __restrict__ xc,     // (B*L, DINNER)
    const float* __restrict__ dbl,    // (B*L, 96): B at +64, C at +80
    const float* __restrict__ A_log,  // (DINNER, DSTATE)
    const float* __restrict__ Dp,     // (DINNER)
    float* __restrict__ xz)           // (B*L, 4096): z at +2048, y out at +0
{
    const int tid = blockIdx.x * blockDim.x + threadIdx.x;
    const int g = tid >> 4;           // channel group over B*DINNER
    const int s = tid & 15;
    if (g >= BSZ * DINNER) return;
    const int b = g / DINNER;
    const int d = g - b * DINNER;

    const float As = -expf(A_log[(size_t)d * DSTATE + s]);
    const float Dd = Dp[d];
    float h = 0.f;
    const size_t rowbase = (size_t)b * SEQL;

    for (int l = 0; l < SEQL; ++l) {
        const size_t r = rowbase + l;
        const float dtv = dt[r * DINNER + d];
        const float xcv = xc[r * DINNER + d];
        const float Bs  = dbl[r * 96 + DTRANK + s];
        const float Cs  = dbl[r * 96 + DTRANK + DSTATE + s];
        const float dA  = expf(dtv * As);
        h = dA * h + dtv * Bs * xcv;
        float p = h * Cs;
        // reduce over the 16-lane state group (xor stays inside the group)
        p += __shfl_xor(p, 1);
        p += __shfl_xor(p, 2);
        p += __shfl_xor(p, 4);
        p += __shfl_xor(p, 8);
        if (s == 0) {
            const float zv = xz[r * (2 * DINNER) + DINNER + d];
            float y = p + xcv * Dd;
            y *= zv / (1.f + expf(-zv));          // * silu(z)
            xz[r * (2 * DINNER) + d] = y;
        }
    }
}

// ---------------------------------------------------------------------------
// Host launcher
// ---------------------------------------------------------------------------
template<int JT>
static inline void launch_gemm_part(const float* A, long lda,
                                    const float* W, long ldw,
                                    float* C, long ldc, const float* bias,
                                    int M, int K, int act,
                                    int ntile0, int nstrips, hipStream_t stream)
{
    if (nstrips <= 0) return;
    const int mtiles  = M >> 4;
    const int mgroups = ((mtiles >> 1) + 7) / 8;  // 2 m-tiles/wave, 8 waves/block
    const int blocks  = nstrips * mgroups;
    const float* Wp = W + (size_t)(ntile0 * 16) * ldw;
    float*       Cp = C + ntile0 * 16;
    const float* bp = bias ? bias + ntile0 * 16 : nullptr;
    wmma_gemm_f32<JT><<<blocks, 256, 0, stream>>>(
        A, lda, Wp, ldw, Cp, ldc, bp, mtiles, nstrips, K, act);
}

static inline void launch_gemm(const float* A, long lda, const float* W, long ldw,
                               float* C, long ldc, const float* bias,
                               int M, int N, int K, int act, hipStream_t stream)
{
    const int ntiles = N >> 4;
    const int full4  = ntiles >> 2;               // strips of 4 tiles
    launch_gemm_part<4>(A, lda, W, ldw, C, ldc, bias, M, K, act,
                        0, full4, stream);
    int done = full4 << 2;
    if (ntiles - done >= 2) {                     // strip of 2 tiles
        launch_gemm_part<2>(A, lda, W, ldw, C, ldc, bias, M, K, act,
                            done, 1, stream);
        done += 2;
    }
    if (ntiles - done >= 1) {                     // final single tile
        launch_gemm_part<1>(A, lda, W, ldw, C, ldc, bias, M, K, act,
                            done, 1, stream);
    }
}

extern "C" void kernel_launch(void* const* d_in, const int* in_sizes, int n_in,
                              void* d_out, int out_size, void* d_ws, size_t ws_size,
                              hipStream_t stream)
{
    const float* x_input    = (const float*)d_in[0];
    const float* in_proj_w  = (const float*)d_in[1];
    const float* conv_w     = (const float*)d_in[2];
    const float* conv_b     = (const float*)d_in[3];
    const float* x_proj_w   = (const float*)d_in[4];
    const float* dt_proj_w  = (const float*)d_in[5];
    const float* dt_proj_b  = (const float*)d_in[6];
    const float* A_log      = (const float*)d_in[7];
    const float* Dp         = (const float*)d_in[8];
    const float* out_proj_w = (const float*)d_in[9];
    float* out = (float*)d_out;

    const int M = BSZ * SEQL;                 // 2048 token rows

    // workspace layout (floats), ~76 MB total
    float* ws    = (float*)d_ws;
    float* xz    = ws;                                        // (M, 4096)
    float* xc    = xz  + (size_t)M * 2 * DINNER;              // (M, 2048)
    float* dbl   = xc  + (size_t)M * DINNER;                  // (M, 96)
    float* dtbuf = dbl + (size_t)M * (DTRANK + 2 * DSTATE);   // (M, 2048)
    float* x_mid = dtbuf + (size_t)M * DINNER;                // (M, 1024)

    for (int layer = 0; layer < NLAYER; ++layer) {
        const float* xin = (layer == 0) ? x_input : x_mid;
        float* dst       = (layer == NLAYER - 1) ? out : x_mid;

        const float* in_w = in_proj_w  + (size_t)layer * (2 * DINNER) * DMODEL;
        const float* cw   = conv_w     + (size_t)layer * DINNER * 4;
        const float* cb   = conv_b     + (size_t)layer * DINNER;
        const float* xw   = x_proj_w   + (size_t)layer * (DTRANK + 2 * DSTATE) * DINNER;
        const float* dtw  = dt_proj_w  + (size_t)layer * DINNER * DTRANK;
        const float* dtb  = dt_proj_b  + (size_t)layer * DINNER;
        const float* Al   = A_log      + (size_t)layer * DINNER * DSTATE;
        const float* Dl   = Dp         + (size_t)layer * DINNER;
        const float* ow   = out_proj_w + (size_t)layer * DMODEL * DINNER;

        // 1) xz = x @ in_w^T                       (M x 4096, K=1024)
        launch_gemm(xin, DMODEL, in_w, DMODEL, xz, 2 * DINNER, nullptr,
                    M, 2 * DINNER, DMODEL, 0, stream);

        // 2) xc = silu(causal_depthwise_conv(xr) + cb)
        {
            const int total = M * DINNER;
            conv_silu_kernel<<<(total + 255) / 256, 256, 0, stream>>>(xz, cw, cb, xc);
        }

        // 3) dbl = xc @ xw^T                        (M x 96, K=2048)
        launch_gemm(xc, DINNER, xw, DINNER, dbl, DTRANK + 2 * DSTATE, nullptr,
                    M, DTRANK + 2 * DSTATE, DINNER, 0, stream);

        // 4) dt = softplus(dbl[:, :64] @ dtw^T + dtb)   (M x 2048, K=64)
        launch_gemm(dbl, DTRANK + 2 * DSTATE, dtw, DTRANK, dtbuf, DINNER, dtb,
                    M, DINNER, DTRANK, 1, stream);

        // 5) selective scan + skip + gate; y written into xr half of xz
        {
            const int total = BSZ * DINNER * DSTATE;   // 16 lanes per channel
            mamba_scan_kernel<<<(total + 255) / 256, 256, 0, stream>>>(
                dtbuf, xc, dbl, Al, Dl, xz);
        }

        // 6) dst = y @ ow^T                          (M x 1024, K=2048)
        launch_gemm(xz, 2 * DINNER, ow, DINNER, dst, DMODEL, nullptr,
                    M, DMODEL, DINNER, 0, stream);
    }
}